// GraphAttentionLayer_17712445129455
// MI455X (gfx1250) — compile-verified
//
#include <hip/hip_runtime.h>

#define ALPHA 0.2f

typedef __attribute__((ext_vector_type(2))) float v2f;
typedef __attribute__((ext_vector_type(8))) float v8f;

__device__ __forceinline__ v8f wmma_f32_16x16x4(v2f a, v2f b, v8f c) {
  // (neg_a, A, neg_b, B, c_mod, C, reuse_a, reuse_b)
  return __builtin_amdgcn_wmma_f32_16x16x4_f32(false, a, false, b, (short)0, c,
                                               false, false);
}

__device__ __forceinline__ float leaky(float t) {
  return (t >= 0.0f) ? t : ALPHA * t;
}

// -------------------- K1: h = x @ W^T  (16384x128 = 16384x128 * 128x128) ---
// block = 128 threads = 4 waves; wave computes 16 rows x 128 cols.
// f32 WMMA 16x16x4 A-layout: lane L(<16): M=L, K={0,1}; lane L+16: M=L, K={2,3}
__global__ void __launch_bounds__(128) k_h_gemm(const float* __restrict__ x,
                                                const float* __restrict__ W,
                                                float* __restrict__ h) {
  const int lane = threadIdx.x & 31;
  const int wave = threadIdx.x >> 5;
  const int rl = lane & 15;
  const int hi = lane >> 4;
  const int i0 = blockIdx.x * 64 + wave * 16;

  v8f acc[8];
  v8f zero = {};
#pragma unroll
  for (int t = 0; t < 8; ++t) acc[t] = zero;

  const float* xrow = x + (size_t)(i0 + rl) * 128 + 2 * hi;
#pragma unroll 4
  for (int k = 0; k < 128; k += 4) {
    v2f afrag = *(const v2f*)(xrow + k);  // x[i0+rl][k+2hi .. +1]
#pragma unroll
    for (int ot = 0; ot < 8; ++ot) {
      // B[K][N] = W[ot*16+N][k+K]; lane n: N=n, K={0,1}; lane n+16: K={2,3}
      v2f bfrag = *(const v2f*)(W + (size_t)(ot * 16 + rl) * 128 + k + 2 * hi);
      acc[ot] = wmma_f32_16x16x4(afrag, bfrag, acc[ot]);
    }
  }
  // C/D layout: VGPR r: lanes 0-15 -> row r, lanes 16-31 -> row r+8; col = rl
#pragma unroll
  for (int ot = 0; ot < 8; ++ot)
#pragma unroll
    for (int r = 0; r < 8; ++r)
      h[(size_t)(i0 + r + hi * 8) * 128 + ot * 16 + rl] = acc[ot][r];
}

// -------------------- K2: si = h.a1, sj = h.a2 (one wave per row) ----------
__global__ void __launch_bounds__(256) k_scores(const float* __restrict__ h,
                                                const float* __restrict__ a,
                                                float* __restrict__ si,
                                                float* __restrict__ sj) {
  const int row = blockIdx.x * 8 + (threadIdx.x >> 5);
  const int lane = threadIdx.x & 31;
  float psi = 0.0f, psj = 0.0f;
#pragma unroll
  for (int c = lane; c < 128; c += 32) {
    float hv = h[(size_t)row * 128 + c];
    psi += hv * a[c];
    psj += hv * a[128 + c];
  }
#pragma unroll
  for (int off = 16; off > 0; off >>= 1) {
    psi += __shfl_xor(psi, off, 32);
    psj += __shfl_xor(psj, off, 32);
  }
  if (lane == 0) {
    si[row] = psi;
    sj[row] = psj;
  }
}

// -------------------- K3: per-batch max of sj ------------------------------
__global__ void __launch_bounds__(256) k_maxsj(const float* __restrict__ sj,
                                               float* __restrict__ mx) {
  __shared__ float red[256];
  const int b = blockIdx.x;
  float m = -3.402823466e38f;
  for (int j = threadIdx.x; j < 2048; j += 256)
    m = fmaxf(m, sj[(size_t)b * 2048 + j]);
  red[threadIdx.x] = m;
  __syncthreads();
  for (int s = 128; s > 0; s >>= 1) {
    if (threadIdx.x < s) red[threadIdx.x] = fmaxf(red[threadIdx.x], red[threadIdx.x + s]);
    __syncthreads();
  }
  if (threadIdx.x == 0) mx[b] = red[0];
}

// -------------------- K4: out = softmax(leaky(si+sj)) @ h  -----------------
// Fused flash-style GAT: attention never materialized (saves 134 MB of HBM
// traffic). Monotonic LeakyReLU => row max = leaky(si + max_j sj), enabling a
// single pass. P tile generated in registers as the WMMA A operand; Z
// accumulated in the same sweep.
//
// LDS layout: j-pair interleaved with 288-float pair stride:
//   hs2[(j>>1)*288 + 2*c + (j&1)] = h[j][c]
//  -> B fragment {h[jc][c], h[jc+1][c]} is ONE aligned ds_load_b64, and since
//     288 mod 64 = 32, the two half-waves land on disjoint bank halves
//     (conflict-free full-wave b64 reads).
// Software pipeline: while computing chunk n from LDS, global b128 loads for
// chunk n+1 sit in registers; after the read barrier they are written as two
// coalesced ds_store_b128 per work item.
#define PSTRIDE 288  // 256 data + 32 pad floats per j-pair
__global__ void __launch_bounds__(128) k_attn(const float* __restrict__ h,
                                              const float* __restrict__ si,
                                              const float* __restrict__ sj,
                                              const float* __restrict__ mx,
                                              float* __restrict__ out) {
  __shared__ __align__(16) float hs2[32 * PSTRIDE];  // 36 KB
  __shared__ __align__(16) float sjs[64];
  __shared__ float zbuf[4 * 16];

  const int tid = threadIdx.x;
  const int lane = tid & 31;
  const int wave = tid >> 5;
  const int rl = lane & 15;
  const int hi = lane >> 4;
  const int b = blockIdx.y;
  const int i0 = blockIdx.x * 64 + wave * 16;

  const float* hb = h + (size_t)b * 2048 * 128;
  const float* sjb = sj + (size_t)b * 2048;

  const float si_row = si[(size_t)b * 2048 + i0 + rl];
  const float m_row = leaky(si_row + mx[b]);  // exact row max of e

  v8f acc[8];
  v8f zero = {};
#pragma unroll
  for (int t = 0; t < 8; ++t) acc[t] = zero;
  float zacc = 0.0f;

  float4 rA[8], rB[8];
  float sjr = 0.0f;

  // ---- preload + stage chunk 0 ----
#pragma unroll
  for (int k = 0; k < 8; ++k) {
    const int q = tid + 128 * k;
    const int jp = q >> 5, c0 = (q & 31) * 4;
    rA[k] = *(const float4*)(hb + (size_t)(2 * jp) * 128 + c0);
    rB[k] = *(const float4*)(hb + (size_t)(2 * jp + 1) * 128 + c0);
  }
  if (tid < 64) sjr = sjb[tid];
#pragma unroll
  for (int k = 0; k < 8; ++k) {
    const int q = tid + 128 * k;
    const int jp = q >> 5, c0 = (q & 31) * 4;
    float4* dst = (float4*)&hs2[jp * PSTRIDE + c0 * 2];
    dst[0] = make_float4(rA[k].x, rB[k].x, rA[k].y, rB[k].y);
    dst[1] = make_float4(rA[k].z, rB[k].z, rA[k].w, rB[k].w);
  }
  if (tid < 64) sjs[tid] = sjr;
  __syncthreads();

  for (int ch = 0; ch < 32; ++ch) {
    // ---- issue global loads for next chunk into registers (overlap) ----
    if (ch + 1 < 32) {
      const int jn = (ch + 1) * 64;
#pragma unroll
      for (int k = 0; k < 8; ++k) {
        const int q = tid + 128 * k;
        const int jp = q >> 5, c0 = (q & 31) * 4;
        rA[k] = *(const float4*)(hb + (size_t)(jn + 2 * jp) * 128 + c0);
        rB[k] = *(const float4*)(hb + (size_t)(jn + 2 * jp + 1) * 128 + c0);
      }
      if (tid < 64) sjr = sjb[jn + tid];
    }

    // ---- compute on current chunk ----
#pragma unroll 4
    for (int j = 0; j < 64; j += 4) {
      const int jp = (j >> 1) + hi;            // jc = j + 2*hi = 2*jp (even)
      v2f sp = *(const v2f*)&sjs[j + 2 * hi];  // {sj[jc], sj[jc+1]}
      float p0 = __expf(leaky(si_row + sp[0]) - m_row);
      float p1 = __expf(leaky(si_row + sp[1]) - m_row);
      zacc += p0 + p1;
      v2f afrag = {p0, p1};  // P[row rl][jc], P[row rl][jc+1]
      const float* bb = &hs2[jp * PSTRIDE + rl * 2];
#pragma unroll
      for (int ot = 0; ot < 8; ++ot) {
        v2f bfrag = *(const v2f*)(bb + ot * 32);  // one ds_load_b64
        acc[ot] = wmma_f32_16x16x4(afrag, bfrag, acc[ot]);
      }
    }
    __syncthreads();  // all reads of current chunk done

    // ---- write next chunk from registers to LDS ----
    if (ch + 1 < 32) {
#pragma unroll
      for (int k = 0; k < 8; ++k) {
        const int q = tid + 128 * k;
        const int jp = q >> 5, c0 = (q & 31) * 4;
        float4* dst = (float4*)&hs2[jp * PSTRIDE + c0 * 2];
        dst[0] = make_float4(rA[k].x, rB[k].x, rA[k].y, rB[k].y);
        dst[1] = make_float4(rA[k].z, rB[k].z, rA[k].w, rB[k].w);
      }
      if (tid < 64) sjs[tid] = sjr;
    }
    __syncthreads();  // next chunk visible
  }

  // Z[row rl] = (lane contrib j%4 in {0,1}) + (partner lane contrib {2,3})
  float zfull = zacc + __shfl_xor(zacc, 16, 32);
  if (hi == 0) zbuf[wave * 16 + rl] = zfull;
  __syncthreads();
  float zinv[8];
#pragma unroll
  for (int r = 0; r < 8; ++r) zinv[r] = 1.0f / zbuf[wave * 16 + r + hi * 8];

#pragma unroll
  for (int ot = 0; ot < 8; ++ot)
#pragma unroll
    for (int r = 0; r < 8; ++r)
      out[((size_t)b * 2048 + i0 + r + hi * 8) * 128 + ot * 16 + rl] =
          acc[ot][r] * zinv[r];
}

extern "C" void kernel_launch(void* const* d_in, const int* in_sizes, int n_in,
                              void* d_out, int out_size, void* d_ws, size_t ws_size,
                              hipStream_t stream) {
  const float* x = (const float*)d_in[0];   // (8,2048,128) f32
  const float* W = (const float*)d_in[1];   // (128,128) f32
  const float* a = (const float*)d_in[2];   // (1,256) f32
  float* out = (float*)d_out;               // (8,2048,128) f32

  float* h  = (float*)d_ws;                 // 16384*128 f32 = 8 MB
  float* si = h + (size_t)16384 * 128;      // 16384 f32
  float* sj = si + 16384;                   // 16384 f32
  float* mx = sj + 16384;                   // 8 f32

  k_h_gemm<<<dim3(256), dim3(128), 0, stream>>>(x, W, h);
  k_scores<<<dim3(2048), dim3(256), 0, stream>>>(h, a, si, sj);
  k_maxsj<<<dim3(8), dim3(256), 0, stream>>>(sj, mx);
  k_attn<<<dim3(32, 8), dim3(128), 0, stream>>>(h, si, sj, mx, out);
}